// GNN2Seq_50749333570253
// MI455X (gfx1250) — compile-verified
//
#include <hip/hip_runtime.h>

#define N_NODES 50000
#define N_EDGES 800000
#define D_IN    16
#define HDIM    128
#define ED_DIM  16
#define OUT_DIM 20
#define G_SEG   64
#define N_STEPS 5
#define NEG_SLOPE 0.2f

typedef __attribute__((ext_vector_type(2))) float v2f;
typedef __attribute__((ext_vector_type(8))) float v8f;

__device__ __forceinline__ float lrelu(float x) { return x >= 0.f ? x : NEG_SLOPE * x; }

// Monotone float -> uint mapping for atomic max on floats.
__device__ __forceinline__ unsigned ordf(float f) {
  unsigned u = __float_as_uint(f);
  return (u & 0x80000000u) ? ~u : (u | 0x80000000u);
}
__device__ __forceinline__ float unordf(unsigned u) {
  return (u & 0x80000000u) ? __uint_as_float(u & 0x7FFFFFFFu) : __uint_as_float(~u);
}
__device__ __forceinline__ float wave_sum32(float p) {
#pragma unroll
  for (int o = 16; o > 0; o >>= 1) p += __shfl_xor(p, o, 32);
  return p;
}

// ---------------------------------------------------------------------------
// C[M,Nc] = A[M,K] @ W[K,Nc] + bias (optional relu), fp32 WMMA 16x16x4.
// One wave computes a 16x16 tile; 8 waves/block cover 16 rows x up to 128 cols.
// A layout: lane L (m=L&15, half=L>>4) holds A[m][k0+2*half], A[m][k0+2*half+1].
// B layout: lane L (n=L&15, half=L>>4) holds W[k0+2*half][n], W[k0+2*half+1][n].
// C layout: VGPR v, lane L -> row v+8*half, col n.
// ---------------------------------------------------------------------------
__global__ __launch_bounds__(256) void k_gemm_wmma(
    const float* __restrict__ A, const float* __restrict__ W,
    const float* __restrict__ bias, float* __restrict__ C,
    int M, int K, int Nc, int do_relu) {
  __shared__ float As[16][132];  // pitch 132 -> rows land on distinct banks
  const int rowbase = blockIdx.x * 16;
  const int tid = threadIdx.x;
  for (int idx = tid; idx < 16 * K; idx += 256) {
    int r = idx / K, c = idx - r * K;
    As[r][c] = A[(long long)(rowbase + r) * K + c];
  }
  __syncthreads();

  const int wave = tid >> 5, lane = tid & 31;
  const int colbase = wave * 16;
  if (colbase >= Nc) return;
  const int half = lane >> 4;
  const int mr = lane & 15;  // row for A, col for B/C

  v8f acc = {0.f, 0.f, 0.f, 0.f, 0.f, 0.f, 0.f, 0.f};
  for (int k0 = 0; k0 < K; k0 += 4) {
    v2f a, b;
    a.x = As[mr][k0 + 2 * half];
    a.y = As[mr][k0 + 2 * half + 1];
    const float* wp = W + (long long)(k0 + 2 * half) * Nc + colbase + mr;
    b.x = wp[0];
    b.y = wp[Nc];
    acc = __builtin_amdgcn_wmma_f32_16x16x4_f32(false, a, false, b, (short)0,
                                                acc, false, false);
  }
#pragma unroll
  for (int v = 0; v < 8; v++) {
    int row = rowbase + v + 8 * half;
    int col = colbase + mr;
    if (row < M) {
      float val = acc[v] + bias[col];
      if (do_relu) val = fmaxf(val, 0.f);
      C[(long long)row * Nc + col] = val;
    }
  }
}

// nan_to_num
__global__ __launch_bounds__(256) void k_nan_clean(const float* __restrict__ x,
                                                   float* __restrict__ y, int n) {
  int i = blockIdx.x * 256 + threadIdx.x;
  if (i >= n) return;
  float v = x[i];
  if (isnan(v)) v = 0.f;
  else if (isinf(v)) v = v > 0.f ? 3.4028234663852886e38f : -3.4028234663852886e38f;
  y[i] = v;
}

__global__ __launch_bounds__(256) void k_relu(float* __restrict__ x, int n) {
  int i = blockIdx.x * 256 + threadIdx.x;
  if (i < n) x[i] = fmaxf(x[i], 0.f);
}

// Edge pass 1 (one wave per edge, 4 cols/lane):
//   ea = edge_attr[e] @ We (on the fly); easum[dst] += ea; deg[dst] += 1;
//   score = att . leaky_relu(xl[src]+xr[dst]+ea); segment max into m_ord[dst].
__global__ __launch_bounds__(256) void k_edge_pass1(
    const int* __restrict__ src, const int* __restrict__ dst,
    const float* __restrict__ eattr, const float* __restrict__ We,
    const float* __restrict__ att, const float* __restrict__ xl,
    const float* __restrict__ xr, float* __restrict__ easum,
    float* __restrict__ deg, unsigned* __restrict__ m_ord,
    float* __restrict__ scoreE, int nE) {
  int wid = (blockIdx.x * 256 + threadIdx.x) >> 5;
  int lane = threadIdx.x & 31;
  if (wid >= nE) return;
  int s = src[wid], d = dst[wid];
  int c0 = lane * 4;
  const float* er = eattr + (long long)wid * ED_DIM;
  float e0 = 0.f, e1 = 0.f, e2 = 0.f, e3 = 0.f;
#pragma unroll
  for (int j = 0; j < ED_DIM; j++) {
    float aj = er[j];
    const float* w = We + j * HDIM + c0;
    e0 += aj * w[0]; e1 += aj * w[1]; e2 += aj * w[2]; e3 += aj * w[3];
  }
  float* es = easum + (long long)d * HDIM + c0;
  atomicAdd(es + 0, e0); atomicAdd(es + 1, e1);
  atomicAdd(es + 2, e2); atomicAdd(es + 3, e3);
  const float4 lv = *(const float4*)(xl + (long long)s * HDIM + c0);
  const float4 rv = *(const float4*)(xr + (long long)d * HDIM + c0);
  const float4 av = *(const float4*)(att + c0);
  float p = lrelu(lv.x + rv.x + e0) * av.x + lrelu(lv.y + rv.y + e1) * av.y +
            lrelu(lv.z + rv.z + e2) * av.z + lrelu(lv.w + rv.w + e3) * av.w;
  p = wave_sum32(p);
  if (lane == 0) {
    scoreE[wid] = p;
    atomicMax(m_ord + d, ordf(p));
    atomicAdd(deg + d, 1.f);
  }
}

// Self-loop pass (one wave per node): ea_self = easum/max(deg,1);
// self score; m_final = max(edge max, self score); s = exp(self - m_final).
__global__ __launch_bounds__(256) void k_self_pass(
    const float* __restrict__ xl, const float* __restrict__ xr,
    const float* __restrict__ easum, const float* __restrict__ deg,
    const float* __restrict__ att, const unsigned* __restrict__ m_ord,
    float* __restrict__ m_final, float* __restrict__ scoreSelf,
    float* __restrict__ sArr, int nN) {
  int n = (blockIdx.x * 256 + threadIdx.x) >> 5;
  int lane = threadIdx.x & 31;
  if (n >= nN) return;
  int c0 = lane * 4;
  float dv = fmaxf(deg[n], 1.f);
  const float4 ev = *(const float4*)(easum + (long long)n * HDIM + c0);
  const float4 lv = *(const float4*)(xl + (long long)n * HDIM + c0);
  const float4 rv = *(const float4*)(xr + (long long)n * HDIM + c0);
  const float4 av = *(const float4*)(att + c0);
  float p = lrelu(lv.x + rv.x + ev.x / dv) * av.x +
            lrelu(lv.y + rv.y + ev.y / dv) * av.y +
            lrelu(lv.z + rv.z + ev.z / dv) * av.z +
            lrelu(lv.w + rv.w + ev.w / dv) * av.w;
  p = wave_sum32(p);
  if (lane == 0) {
    float mf = fmaxf(unordf(m_ord[n]), p);  // fmaxf drops NaN from empty init
    m_final[n] = mf;
    scoreSelf[n] = p;
    sArr[n] = __expf(p - mf);
  }
}

// Edge pass 2 (thread per edge): a = exp(score - m[dst]); s[dst] += a.
__global__ __launch_bounds__(256) void k_edge_pass2(
    const int* __restrict__ dst, float* __restrict__ scoreE,
    const float* __restrict__ m_final, float* __restrict__ sArr, int nE) {
  int e = blockIdx.x * 256 + threadIdx.x;
  if (e >= nE) return;
  int d = dst[e];
  float a = __expf(scoreE[e] - m_final[d]);
  scoreE[e] = a;
  atomicAdd(sArr + d, a);
}

// Node init (wave per node): out = bias + xl * alpha_self.
__global__ __launch_bounds__(256) void k_node_init(
    const float* __restrict__ xl, const float* __restrict__ bias,
    const float* __restrict__ scoreSelf, const float* __restrict__ m_final,
    const float* __restrict__ sArr, float* __restrict__ out, int nN) {
  int n = (blockIdx.x * 256 + threadIdx.x) >> 5;
  int lane = threadIdx.x & 31;
  if (n >= nN) return;
  int c0 = lane * 4;
  float as = __expf(scoreSelf[n] - m_final[n]) / fmaxf(sArr[n], 1e-16f);
  const float4 lv = *(const float4*)(xl + (long long)n * HDIM + c0);
  const float4 bv = *(const float4*)(bias + c0);
  float4 o;
  o.x = bv.x + lv.x * as; o.y = bv.y + lv.y * as;
  o.z = bv.z + lv.z * as; o.w = bv.w + lv.w * as;
  *(float4*)(out + (long long)n * HDIM + c0) = o;
}

// Edge pass 3 (wave per edge): out[dst] += xl[src] * alpha.
__global__ __launch_bounds__(256) void k_edge_pass3(
    const int* __restrict__ src, const int* __restrict__ dst,
    const float* __restrict__ xl, const float* __restrict__ scoreE,
    const float* __restrict__ sArr, float* __restrict__ out, int nE) {
  int wid = (blockIdx.x * 256 + threadIdx.x) >> 5;
  int lane = threadIdx.x & 31;
  if (wid >= nE) return;
  int s = src[wid], d = dst[wid];
  float al = scoreE[wid] / fmaxf(sArr[d], 1e-16f);
  int c0 = lane * 4;
  const float4 lv = *(const float4*)(xl + (long long)s * HDIM + c0);
  float* op = out + (long long)d * HDIM + c0;
  atomicAdd(op + 0, lv.x * al); atomicAdd(op + 1, lv.y * al);
  atomicAdd(op + 2, lv.z * al); atomicAdd(op + 3, lv.w * al);
}

// gate[n] = dot(h[n], W2) + b2 (wave per node).
__global__ __launch_bounds__(256) void k_gate2(
    const float* __restrict__ h, const float* __restrict__ W2,
    const float* __restrict__ b2, float* __restrict__ gate, int nN) {
  int n = (blockIdx.x * 256 + threadIdx.x) >> 5;
  int lane = threadIdx.x & 31;
  if (n >= nN) return;
  int c0 = lane * 4;
  const float4 hv = *(const float4*)(h + (long long)n * HDIM + c0);
  const float4 wv = *(const float4*)(W2 + c0);
  float p = hv.x * wv.x + hv.y * wv.y + hv.z * wv.z + hv.w * wv.w;
  p = wave_sum32(p);
  if (lane == 0) gate[n] = p + b2[0];
}

__global__ __launch_bounds__(256) void k_gseg_max(
    const float* __restrict__ gate, const int* __restrict__ batch,
    unsigned* __restrict__ gm_ord, int nN) {
  int n = blockIdx.x * 256 + threadIdx.x;
  if (n >= nN) return;
  atomicMax(gm_ord + batch[n], ordf(gate[n]));
}

__global__ __launch_bounds__(256) void k_gseg_exp(
    float* __restrict__ gate, const int* __restrict__ batch,
    const unsigned* __restrict__ gm_ord, float* __restrict__ gs, int nN) {
  int n = blockIdx.x * 256 + threadIdx.x;
  if (n >= nN) return;
  int b = batch[n];
  float a = __expf(gate[n] - unordf(gm_ord[b]));
  gate[n] = a;
  atomicAdd(gs + b, a);
}

// g[batch[n]] += h[n] * (a[n]/max(gs,1e-16))  (wave per node)
__global__ __launch_bounds__(256) void k_gacc(
    const float* __restrict__ h, const float* __restrict__ gate,
    const int* __restrict__ batch, const float* __restrict__ gs,
    float* __restrict__ g, int nN) {
  int n = (blockIdx.x * 256 + threadIdx.x) >> 5;
  int lane = threadIdx.x & 31;
  if (n >= nN) return;
  int b = batch[n];
  float w = gate[n] / fmaxf(gs[b], 1e-16f);
  int c0 = lane * 4;
  const float4 hv = *(const float4*)(h + (long long)n * HDIM + c0);
  float* gp = g + (long long)b * HDIM + c0;
  atomicAdd(gp + 0, hv.x * w); atomicAdd(gp + 1, hv.y * w);
  atomicAdd(gp + 2, hv.z * w); atomicAdd(gp + 3, hv.w * w);
}

// out[r,c] = b[c] + sum_k gh1[r,k]*W[k,c]   (64x20, thread per element)
__global__ __launch_bounds__(256) void k_out2(
    const float* __restrict__ gh1, const float* __restrict__ W,
    const float* __restrict__ b, float* __restrict__ out) {
  int idx = blockIdx.x * 256 + threadIdx.x;
  if (idx >= G_SEG * OUT_DIM) return;
  int r = idx / OUT_DIM, c = idx - r * OUT_DIM;
  float acc = b[c];
#pragma unroll
  for (int k = 0; k < 64; k++) acc += gh1[r * 64 + k] * W[k * OUT_DIM + c];
  out[idx] = acc;
}

extern "C" void kernel_launch(void* const* d_in, const int* in_sizes, int n_in,
                              void* d_out, int out_size, void* d_ws, size_t ws_size,
                              hipStream_t stream) {
  const float* x    = (const float*)d_in[0];
  const int* eidx   = (const int*)d_in[1];
  const int* src    = eidx;
  const int* dst    = eidx + N_EDGES;
  const float* eatt = (const float*)d_in[2];
  const int* batch  = (const int*)d_in[3];
  // d_in[4] = n_steps (== 5)
  auto P = [&](int i) { return (const float*)d_in[i]; };
  // JAX tree-flatten (sorted keys): enc[5..25], ga_nn[26,27], gate1[28,29],
  // gate2[30,31], next1[32,33], next2[34,35], out1[36,37], out2[38,39], prop[40..60].
  // Each GAT: We, att, bias, lin_l.W, lin_l.b, lin_r.W, lin_r.b.

  float* ws = (float*)d_ws;
  const long long NH = (long long)N_NODES * HDIM;
  float* s0 = ws;
  float* s1 = ws + NH;
  float* xl = ws + 2 * NH;
  float* xr = ws + 3 * NH;
  float* easum = ws + 4 * NH;
  float* tmp1 = ws + 5 * NH;
  float* nextst = ws + 6 * NH;
  float* scoreE = ws + 7 * NH;
  float* scoreSelf = scoreE + N_EDGES;
  float* m_final = scoreSelf + N_NODES;
  float* sArr = m_final + N_NODES;
  float* deg = sArr + N_NODES;
  float* gate = deg + N_NODES;
  unsigned* m_ord = (unsigned*)(gate + N_NODES);
  unsigned* gm_ord = m_ord + N_NODES;
  float* gs = (float*)(gm_ord + G_SEG);
  float* gbuf = gs + G_SEG;
  float* gh1 = gbuf + (long long)G_SEG * HDIM;
  float* outp = (float*)d_out;

  auto gemm = [&](const float* A, const float* W, const float* bias, float* C,
                  int M, int K, int Nc, int relu) {
    hipLaunchKernelGGL(k_gemm_wmma, dim3(M / 16), dim3(256), 0, stream,
                       A, W, bias, C, M, K, Nc, relu);
  };

  auto gat_layer = [&](const float* in, float* out, int base, int din, int relu) {
    const float* We = P(base), *att = P(base + 1), *bias = P(base + 2);
    const float* lW = P(base + 3), *lb = P(base + 4);
    const float* rW = P(base + 5), *rb = P(base + 6);
    gemm(in, lW, lb, xl, N_NODES, din, HDIM, 0);
    gemm(in, rW, rb, xr, N_NODES, din, HDIM, 0);
    hipMemsetAsync(easum, 0, NH * sizeof(float), stream);
    hipMemsetAsync(deg, 0, N_NODES * sizeof(float), stream);
    hipMemsetAsync(m_ord, 0, N_NODES * sizeof(unsigned), stream);
    hipLaunchKernelGGL(k_edge_pass1, dim3((N_EDGES + 7) / 8), dim3(256), 0, stream,
                       src, dst, eatt, We, att, xl, xr, easum, deg, m_ord, scoreE, N_EDGES);
    hipLaunchKernelGGL(k_self_pass, dim3((N_NODES + 7) / 8), dim3(256), 0, stream,
                       xl, xr, easum, deg, att, m_ord, m_final, scoreSelf, sArr, N_NODES);
    hipLaunchKernelGGL(k_edge_pass2, dim3((N_EDGES + 255) / 256), dim3(256), 0, stream,
                       dst, scoreE, m_final, sArr, N_EDGES);
    hipLaunchKernelGGL(k_node_init, dim3((N_NODES + 7) / 8), dim3(256), 0, stream,
                       xl, bias, scoreSelf, m_final, sArr, out, N_NODES);
    hipLaunchKernelGGL(k_edge_pass3, dim3((N_EDGES + 7) / 8), dim3(256), 0, stream,
                       src, dst, xl, scoreE, sArr, out, N_EDGES);
    if (relu)
      hipLaunchKernelGGL(k_relu, dim3((int)((NH + 255) / 256)), dim3(256), 0, stream,
                         out, (int)NH);
  };

  // state = nan_to_num(x) (N x 16), staged in tmp1
  hipLaunchKernelGGL(k_nan_clean, dim3((N_NODES * D_IN + 255) / 256), dim3(256), 0,
                     stream, x, tmp1, N_NODES * D_IN);
  // Encoder stack
  gat_layer(tmp1, s0, 5, D_IN, 1);
  gat_layer(s0, s1, 12, HDIM, 1);
  gat_layer(s1, s0, 19, HDIM, 0);
  float* cur = s0;

  for (int step = 0; step < N_STEPS; step++) {
    float* o1 = (cur == s0) ? s1 : s0;
    gat_layer(cur, o1, 40, HDIM, 1);
    float* o2 = (o1 == s0) ? s1 : s0;
    gat_layer(o1, o2, 47, HDIM, 1);
    float* o3 = (o2 == s0) ? s1 : s0;
    gat_layer(o2, o3, 54, HDIM, 0);
    cur = o3;

    // next_state = next2(relu(next1(state)))
    gemm(cur, P(32), P(33), tmp1, N_NODES, HDIM, HDIM, 1);
    gemm(tmp1, P(34), P(35), nextst, N_NODES, HDIM, HDIM, 0);

    // global attention
    gemm(cur, P(28), P(29), tmp1, N_NODES, HDIM, HDIM, 1);  // gate1 + relu
    hipLaunchKernelGGL(k_gate2, dim3((N_NODES + 7) / 8), dim3(256), 0, stream,
                       tmp1, P(30), P(31), gate, N_NODES);
    hipMemsetAsync(gm_ord, 0, G_SEG * sizeof(unsigned), stream);
    hipMemsetAsync(gs, 0, G_SEG * sizeof(float), stream);
    hipMemsetAsync(gbuf, 0, (size_t)G_SEG * HDIM * sizeof(float), stream);
    hipLaunchKernelGGL(k_gseg_max, dim3((N_NODES + 255) / 256), dim3(256), 0, stream,
                       gate, batch, gm_ord, N_NODES);
    hipLaunchKernelGGL(k_gseg_exp, dim3((N_NODES + 255) / 256), dim3(256), 0, stream,
                       gate, batch, gm_ord, gs, N_NODES);
    gemm(cur, P(26), P(27), tmp1, N_NODES, HDIM, HDIM, 0);  // ga_nn
    hipLaunchKernelGGL(k_gacc, dim3((N_NODES + 7) / 8), dim3(256), 0, stream,
                       tmp1, gate, batch, gs, gbuf, N_NODES);

    // out head
    gemm(gbuf, P(36), P(37), gh1, G_SEG, HDIM, 64, 1);      // out1 + relu
    hipLaunchKernelGGL(k_out2, dim3((G_SEG * OUT_DIM + 255) / 256), dim3(256), 0,
                       stream, gh1, P(38), P(39), outp + (long long)step * G_SEG * OUT_DIM);

    cur = nextst;  // consumed at the start of next iteration before rewrite
  }
}